// GCN_58858231824590
// MI455X (gfx1250) — compile-verified
//
#include <hip/hip_runtime.h>

typedef __attribute__((ext_vector_type(2))) float v2f;
typedef __attribute__((ext_vector_type(8))) float v8f;

#define NODES 50000
#define HID   256
#define NLAB  128
#define BN_EPS 1e-5f

// ---------------------------------------------------------------------------
// WMMA GEMM: C(MxN) = A(MxK=256) * B(256xN) [+ bias[n]]
// block = 256 threads = 8 waves; each wave computes one 16x16 tile.
// block tile = 16 rows x 128 cols. grid = (ceil(M/16), N/128).
// A strip (16x256 f32 = 16 KB) and B chunks (64x128 f32 = 32 KB) staged in
// LDS; inner loop is pure ds_load + v_wmma_f32_16x16x4_f32 (64 MMAs/tile).
// ---------------------------------------------------------------------------
__global__ void __launch_bounds__(256)
gemm_wmma_f32(const float* __restrict__ A, const float* __restrict__ B,
              const float* __restrict__ bias, float* __restrict__ C,
              int M, int N) {
  __shared__ float As[16 * 256];   // 16 KB
  __shared__ float Bs[64 * 128];   // 32 KB (one 64-row K-chunk of the panel)

  const int tid   = threadIdx.x;
  const int m0    = blockIdx.x * 16;
  const int nbase = blockIdx.y * 128;

  // Cooperative load of 16x256 A strip as float4 (zero-pad rows >= M).
  for (int i = tid; i < 16 * 64; i += 256) {
    const int r  = i >> 6;
    const int c4 = i & 63;
    float4 v = make_float4(0.f, 0.f, 0.f, 0.f);
    const int row = m0 + r;
    if (row < M) v = reinterpret_cast<const float4*>(A + (size_t)row * 256)[c4];
    reinterpret_cast<float4*>(As)[i] = v;
  }

  const int wave = tid >> 5;
  const int lane = tid & 31;
  const int ncol = wave * 16 + (lane & 15);   // column within 128-wide panel
  const int n    = nbase + ncol;              // global output column
  const int kh   = (lane >> 4) << 1;          // 0 (lanes 0-15) / 2 (lanes 16-31)
  const int mrow = lane & 15;

  // C/D layout: VGPR r holds row m0 + r + 8*(lane>>4), column n.
  v8f c;
  const float bv = bias ? bias[n] : 0.f;
#pragma unroll
  for (int r = 0; r < 8; ++r) c[r] = bv;

  for (int kc = 0; kc < 256; kc += 64) {
    __syncthreads();   // As ready (1st iter) / previous Bs chunk consumed
    // Stage 64x128 B chunk: 2048 float4, 8 per thread, fully coalesced.
    for (int i = tid; i < 64 * 32; i += 256) {
      const int r  = i >> 5;
      const int c4 = i & 31;
      reinterpret_cast<float4*>(Bs)[i] =
          reinterpret_cast<const float4*>(B + (size_t)(kc + r) * N + nbase)[c4];
    }
    __syncthreads();

#pragma unroll
    for (int kk = 0; kk < 64; kk += 4) {
      v2f a, b;
      // A 16x4: v0 = K+0 / K+2, v1 = K+1 / K+3 (consecutive pair in LDS)
      a.x = As[mrow * 256 + kc + kk + kh];
      a.y = As[mrow * 256 + kc + kk + kh + 1];
      // B 4x16: rows of B striped across lanes, 512 B apart in LDS
      b.x = Bs[(kk + kh) * 128 + ncol];
      b.y = Bs[(kk + kh + 1) * 128 + ncol];
      c = __builtin_amdgcn_wmma_f32_16x16x4_f32(false, a, false, b,
                                                (short)0, c, false, false);
    }
  }

  const int mbase = m0 + ((lane >> 4) << 3);
  if (m0 + 16 <= M) {           // fast path: whole tile in-bounds (no branches)
#pragma unroll
    for (int r = 0; r < 8; ++r)
      C[(size_t)(mbase + r) * N + n] = c[r];
  } else {                      // only the final partial tile of the last GEMM
#pragma unroll
    for (int r = 0; r < 8; ++r) {
      const int row = mbase + r;
      if (row < M) C[(size_t)row * N + n] = c[r];
    }
  }
}

// out[i*256 + j] = bias[j]  (pre-seed SpMM accumulator with per-column bias)
__global__ void __launch_bounds__(256)
row_bias_init(float* __restrict__ out, const float* __restrict__ bias, size_t total) {
  const size_t t = (size_t)blockIdx.x * 256 + threadIdx.x;
  if (t < total) out[t] = bias[t & 255u];
}

// COO SpMM via atomics: one wave per edge, lane owns 8 consecutive columns.
__global__ void __launch_bounds__(256)
spmm_atomic(const float* __restrict__ ev, const int* __restrict__ er,
            const int* __restrict__ ec, const float* __restrict__ x,
            float* __restrict__ out, int E) {
  const int w    = (int)(((size_t)blockIdx.x * 256 + threadIdx.x) >> 5);
  const int lane = threadIdx.x & 31;
  if (w >= E) return;
  const float v   = ev[w];
  const size_t sb = (size_t)ec[w] * 256;
  const size_t db = (size_t)er[w] * 256;
  const int j = lane << 3;
  const float4 x0 = reinterpret_cast<const float4*>(x + sb + j)[0];
  const float4 x1 = reinterpret_cast<const float4*>(x + sb + j)[1];
  float* o = out + db + j;
  atomicAdd(o + 0, v * x0.x); atomicAdd(o + 1, v * x0.y);
  atomicAdd(o + 2, v * x0.z); atomicAdd(o + 3, v * x0.w);
  atomicAdd(o + 4, v * x1.x); atomicAdd(o + 5, v * x1.y);
  atomicAdd(o + 6, v * x1.z); atomicAdd(o + 7, v * x1.w);
}

__global__ void zero_stats(float* __restrict__ p) {
  p[threadIdx.x] = 0.f;   // launched with 1 block of 512 (sum + sumsq)
}

// Column-wise sum / sum-of-squares: thread = column, block strides rows.
__global__ void __launch_bounds__(256)
bn_stats(const float* __restrict__ x, float* __restrict__ sum,
         float* __restrict__ sq, int Nrows) {
  const int j = threadIdx.x;
  float s = 0.f, q = 0.f;
  for (int i = blockIdx.x; i < Nrows; i += gridDim.x) {
    const float v = x[(size_t)i * 256 + j];
    s += v; q += v * v;
  }
  atomicAdd(&sum[j], s);
  atomicAdd(&sq[j],  q);
}

__global__ void bn_finalize(const float* __restrict__ sum, const float* __restrict__ sq,
                            const float* __restrict__ g, const float* __restrict__ be,
                            float* __restrict__ scale, float* __restrict__ shift,
                            float invN) {
  const int j = threadIdx.x;  // 1 block of 256
  const float m   = sum[j] * invN;
  const float var = sq[j] * invN - m * m;
  const float sc  = g[j] * rsqrtf(var + BN_EPS);
  scale[j] = sc;
  shift[j] = be[j] - m * sc;
}

__global__ void __launch_bounds__(256)
bn_relu(const float* __restrict__ x, const float* __restrict__ scale,
        const float* __restrict__ shift, float* __restrict__ y, size_t total) {
  const size_t t = (size_t)blockIdx.x * 256 + threadIdx.x;
  if (t < total) {
    const int j = (int)(t & 255u);
    const float v = fmaf(x[t], scale[j], shift[j]);
    y[t] = v > 0.f ? v : 0.f;
  }
}

// out[r][:] = x[idx[r]][:]   (float4 per thread)
__global__ void __launch_bounds__(256)
gather_rows(const float* __restrict__ x, const int* __restrict__ idx,
            float* __restrict__ out, int R) {
  const int t  = blockIdx.x * 256 + threadIdx.x;
  const int r  = t >> 6;
  const int c4 = t & 63;
  if (r < R)
    reinterpret_cast<float4*>(out + (size_t)r * 256)[c4] =
        reinterpret_cast<const float4*>(x + (size_t)idx[r] * 256)[c4];
}

extern "C" void kernel_launch(void* const* d_in, const int* in_sizes, int n_in,
                              void* d_out, int out_size, void* d_ws, size_t ws_size,
                              hipStream_t stream) {
  const float* features = (const float*)d_in[0];
  const float* ev       = (const float*)d_in[1];
  const float* W1       = (const float*)d_in[2];
  const float* db1      = (const float*)d_in[3];
  const float* b1       = (const float*)d_in[4];
  const float* g1       = (const float*)d_in[5];
  const float* be1      = (const float*)d_in[6];
  const float* W2       = (const float*)d_in[7];
  const float* b2       = (const float*)d_in[8];
  const float* g2       = (const float*)d_in[9];
  const float* be2      = (const float*)d_in[10];
  const float* Wf       = (const float*)d_in[11];
  const float* bf       = (const float*)d_in[12];
  const int*   er       = (const int*)d_in[13];
  const int*   ec       = (const int*)d_in[14];
  const int*   idx      = (const int*)d_in[15];
  float*       out      = (float*)d_out;

  const int E  = in_sizes[1];
  const int R  = in_sizes[15];           // 25000 gathered rows
  const size_t NH = (size_t)NODES * HID; // 12.8M floats

  float* buf0  = (float*)d_ws;
  float* buf1  = buf0 + NH;
  float* sum   = buf1 + NH;
  float* sq    = sum + HID;
  float* scale = sq + HID;
  float* shift = scale + HID;

  const dim3 blk(256);
  const dim3 gemmGrid((NODES + 15) / 16, HID / 128);          // (3125, 2)
  const dim3 gemmGridF((R + 15) / 16, NLAB / 128);            // (1563, 1)
  const int  ewBlocks = (E * 32 + 255) / 256;                 // wave per edge
  const int  nhBlocks = (int)((NH + 255) / 256);
  const float invN = 1.0f / (float)NODES;

  // ---- Layer 1 ----
  gemm_wmma_f32<<<gemmGrid, blk, 0, stream>>>(features, W1, db1, buf0, NODES, HID);
  row_bias_init<<<nhBlocks, blk, 0, stream>>>(buf1, b1, NH);
  spmm_atomic<<<ewBlocks, blk, 0, stream>>>(ev, er, ec, buf0, buf1, E);
  zero_stats<<<1, 512, 0, stream>>>(sum);
  bn_stats<<<512, blk, 0, stream>>>(buf1, sum, sq, NODES);
  bn_finalize<<<1, 256, 0, stream>>>(sum, sq, g1, be1, scale, shift, invN);
  bn_relu<<<nhBlocks, blk, 0, stream>>>(buf1, scale, shift, buf0, NH);

  // ---- Layer 2 ----
  gemm_wmma_f32<<<gemmGrid, blk, 0, stream>>>(buf0, W2, nullptr, buf1, NODES, HID);
  row_bias_init<<<nhBlocks, blk, 0, stream>>>(buf0, b2, NH);
  spmm_atomic<<<ewBlocks, blk, 0, stream>>>(ev, er, ec, buf1, buf0, E);
  zero_stats<<<1, 512, 0, stream>>>(sum);
  bn_stats<<<512, blk, 0, stream>>>(buf0, sum, sq, NODES);
  bn_finalize<<<1, 256, 0, stream>>>(sum, sq, g2, be2, scale, shift, invN);
  bn_relu<<<nhBlocks, blk, 0, stream>>>(buf0, scale, shift, buf1, NH);

  // ---- Gather then final projection straight into d_out ----
  gather_rows<<<(R * 64 + 255) / 256, blk, 0, stream>>>(buf1, idx, buf0, R);
  gemm_wmma_f32<<<gemmGridF, blk, 0, stream>>>(buf0, Wf, bf, out, R, NLAB);
}